// CircuitBuilder_30382598652135
// MI455X (gfx1250) — compile-verified
//
#include <hip/hip_runtime.h>

#define N_FEATURES 64
#define N_GATES    64
#define N_OUTPUTS  8
#define BASE       (N_FEATURES + 2)        // 66
#define MAX_CONN   (BASE + N_GATES)        // 130
#define BUF_STRIDE 131                     // odd pad -> conflict-free LDS banks
#define BLOCK_THREADS 64                   // 64 samples per block, 2 waves

typedef float v2f __attribute__((ext_vector_type(2)));
typedef float v4f __attribute__((ext_vector_type(4)));
typedef float v8f __attribute__((ext_vector_type(8)));

// ---------------------------------------------------------------------------
// Kernel 1: per-gate top-2 argmax over the first (BASE + g) columns.
// softmax is monotone, so top_k(probs,2) == top-2 argmax of masked logits.
// ---------------------------------------------------------------------------
__global__ void cb_top2_kernel(const float* __restrict__ gw, int* __restrict__ top2) {
    int g = threadIdx.x;
    if (g >= N_GATES) return;
    const int limit = BASE + g;
    const float* row = gw + g * MAX_CONN;
    float m1 = -3.4e38f, m2 = -3.4e38f;
    int i1 = 0, i2 = 0;
    for (int j = 0; j < limit; ++j) {
        float w = row[j];
        if (w > m1) { m2 = m1; i2 = i1; m1 = w; i1 = j; }
        else if (w > m2) { m2 = w; i2 = j; }
    }
    top2[2 * g]     = i1;
    top2[2 * g + 1] = i2;
}

// ---------------------------------------------------------------------------
// Kernel 2: gate-chain in LDS (one sample per lane) + f32 WMMA output GEMM.
// ---------------------------------------------------------------------------
__global__ void __launch_bounds__(BLOCK_THREADS)
cb_circuit_kernel(const float* __restrict__ X,
                  const int*   __restrict__ top2,
                  const float* __restrict__ ow,      // (64, 8)
                  const float* __restrict__ oscale,  // (8,)
                  float*       __restrict__ out)     // (n, 8)
{
    __shared__ float buf[BLOCK_THREADS * BUF_STRIDE];   // per-sample 131-wide buffers
    __shared__ int   t2[2 * N_GATES];
    __shared__ float wlds[N_GATES * N_OUTPUTS];
    __shared__ float slds[N_OUTPUTS];

    const int t = threadIdx.x;
    const long long sample0 = (long long)blockIdx.x * BLOCK_THREADS;

    // ---- Stage X into LDS, coalesced float4 loads (64 samples * 64 feats) ----
    const v4f* Xv = (const v4f*)(X + sample0 * N_FEATURES);
    #pragma unroll
    for (int k = 0; k < 16; ++k) {
        int idx = t + k * BLOCK_THREADS;   // 0..1023 float4 chunks
        int s   = idx >> 4;                // sample within block
        int c   = (idx & 15) << 2;         // feature base
        v4f v = Xv[idx];
        float* dst = &buf[s * BUF_STRIDE + c];
        dst[0] = v.x; dst[1] = v.y; dst[2] = v.z; dst[3] = v.w;
    }
    // constants for each sample's buffer
    buf[t * BUF_STRIDE + N_FEATURES]     = 0.0f;
    buf[t * BUF_STRIDE + N_FEATURES + 1] = 1.0f;
    // wiring + output weights + scale into LDS
    t2[2 * t]     = top2[2 * t];
    t2[2 * t + 1] = top2[2 * t + 1];
    #pragma unroll
    for (int j = 0; j < N_OUTPUTS; ++j)
        wlds[t * N_OUTPUTS + j] = ow[t * N_OUTPUTS + j];
    if (t < N_OUTPUTS) slds[t] = oscale[t];
    __syncthreads();

    // ---- Sequential gate chain: each lane owns one sample's buffer ----
    float* my = &buf[t * BUF_STRIDE];
    #pragma unroll 8
    for (int i = 0; i < N_GATES; ++i) {
        int ca = t2[2 * i], cb = t2[2 * i + 1];
        float a = my[ca];
        float b = my[cb];
        my[BASE + i] = 1.0f - a * b;       // soft-NAND
    }
    __syncthreads();

    // ---- Output GEMM: gates(64 samples x 64) @ W(64 x 16, cols 8..15 = 0) ----
    // V_WMMA_F32_16X16X4_F32, K=64 in 16 steps; each wave does two 16x16 tiles.
    const int lane = t & 31;
    const int wave = t >> 5;
    const int half = lane >> 4;    // ISA: VGPR j holds K = 2*half + j
    const int ln   = lane & 15;    // N column (B/D) or M row (A)

    // Precompute B fragments (shared across both tiles): bw[2*ks + j]
    float bw[32];
    #pragma unroll
    for (int ks = 0; ks < 16; ++ks) {
        #pragma unroll
        for (int j = 0; j < 2; ++j) {
            int k = 4 * ks + 2 * half + j;
            bw[2 * ks + j] = (ln < N_OUTPUTS) ? wlds[k * N_OUTPUTS + ln] : 0.0f;
        }
    }
    float sc = (ln < N_OUTPUTS) ? slds[ln] : 0.0f;

    #pragma unroll
    for (int tile = 0; tile < 2; ++tile) {
        // A fragment: row M = ln of this tile, K = 4*ks + 2*half + j
        const int srow = wave * 32 + tile * 16 + ln;     // sample-in-block
        const float* gv = &buf[srow * BUF_STRIDE + BASE];
        v8f c = {};
        #pragma unroll
        for (int ks = 0; ks < 16; ++ks) {
            v2f a; a.x = gv[4 * ks + 2 * half + 0];
                   a.y = gv[4 * ks + 2 * half + 1];
            v2f b; b.x = bw[2 * ks + 0];
                   b.y = bw[2 * ks + 1];
            c = __builtin_amdgcn_wmma_f32_16x16x4_f32(
                    /*neg_a=*/false, a, /*neg_b=*/false, b,
                    /*c_mod=*/(short)0, c, /*reuse_a=*/false, /*reuse_b=*/false);
        }
        // D layout: VGPR r, lanes 0-15 -> M=r ; lanes 16-31 -> M=8+r ; N=ln
        if (ln < N_OUTPUTS) {
            long long sbase = sample0 + wave * 32 + tile * 16 + 8 * half;
            #pragma unroll
            for (int r = 0; r < 8; ++r) {
                out[(sbase + r) * N_OUTPUTS + ln] = c[r] * sc;
            }
        }
    }
}

// ---------------------------------------------------------------------------
extern "C" void kernel_launch(void* const* d_in, const int* in_sizes, int n_in,
                              void* d_out, int out_size, void* d_ws, size_t ws_size,
                              hipStream_t stream) {
    const float* X  = (const float*)d_in[0];   // (n, 64)
    const float* gw = (const float*)d_in[1];   // (64, 130)
    const float* ow = (const float*)d_in[2];   // (64, 8)
    const float* os = (const float*)d_in[3];   // (8,)
    float* out = (float*)d_out;
    int*   top2 = (int*)d_ws;                  // 128 ints

    const int n = in_sizes[0] / N_FEATURES;    // 524288

    cb_top2_kernel<<<1, 64, 0, stream>>>(gw, top2);

    const int blocks = n / BLOCK_THREADS;      // 8192
    cb_circuit_kernel<<<blocks, BLOCK_THREADS, 0, stream>>>(X, top2, ow, os, out);
}